// WeightConsolidation_70068096467284
// MI455X (gfx1250) — compile-verified
//
#include <hip/hip_runtime.h>
#include <math.h>

typedef float v2f __attribute__((ext_vector_type(2)));
typedef float v4f __attribute__((ext_vector_type(4)));
typedef float v8f __attribute__((ext_vector_type(8)));

#define CONSOLIDATION_RATE 0.001f

// ds_swizzle group-of-32 xor pattern: offset[14:10]=xor_mask, [9:5]=or=0, [4:0]=and=0x1f
template <int XMASK>
__device__ __forceinline__ float swz(float x) {
  return __int_as_float(
      __builtin_amdgcn_ds_swizzle(__float_as_int(x), (XMASK << 10) | 0x1f));
}

__device__ __forceinline__ float waveReduceSum(float v) {
  v += swz<1>(v);
  v += swz<2>(v);
  v += swz<4>(v);
  v += swz<8>(v);
  v += swz<16>(v);
  return v;
}

__device__ __forceinline__ float sigmoidf(float x) {
  return 1.0f / (1.0f + expf(-x));
}

// ---------------------------------------------------------------------------
// Stage 0: zero the workspace accumulators (ws is poisoned by the harness).
// ws layout (floats): [0]=ssq_w [1]=ssq_c [2]=sum_abs [3]=final_strength
//                     [4]=forgetting_strength [8..23]=hidden_bias[16]
// ---------------------------------------------------------------------------
__global__ void wc_init(float* ws) {
  if (threadIdx.x == 0) {
    ws[0] = 0.0f;
    ws[1] = 0.0f;
    ws[2] = 0.0f;
  }
}

// ---------------------------------------------------------------------------
// Stage 1: fused sum-of-squares of current_weights and weight_changes.
// current_weights read with default (RT) hint so it stays in the 192MB L2 for
// stage 3; weight_changes is single-use -> non-temporal load.
// ---------------------------------------------------------------------------
__global__ void __launch_bounds__(256) wc_reduce_sq(
    const v4f* __restrict__ cw4, const v4f* __restrict__ ch4, float* ws,
    int n4) {
  float sw = 0.0f, sc = 0.0f;
  const int stride = (int)(gridDim.x * blockDim.x);
  for (int i = (int)(blockIdx.x * blockDim.x + threadIdx.x); i < n4;
       i += stride) {
    v4f a = cw4[i];
    v4f b = __builtin_nontemporal_load(&ch4[i]);
    sw += a.x * a.x + a.y * a.y + a.z * a.z + a.w * a.w;
    sc += b.x * b.x + b.y * b.y + b.z * b.z + b.w * b.w;
  }
  sw = waveReduceSum(sw);
  sc = waveReduceSum(sc);
  if ((threadIdx.x & 31) == 0) {
    atomicAdd(&ws[0], sw);
    atomicAdd(&ws[1], sc);
  }
}

// ---------------------------------------------------------------------------
// Stage 2: all scalar math (single wave). Computes final_strength,
// change_magnitude, forgetting gate MLP, and the folded 16-wide hidden bias.
// ---------------------------------------------------------------------------
__global__ void wc_scalars(const float* __restrict__ urw1,
                           const float* __restrict__ urb1,
                           const float* __restrict__ fgw1,
                           const float* __restrict__ fgb1,
                           const float* __restrict__ fgw2,
                           const float* __restrict__ fgb2, float* ws,
                           float* out_scalars) {
  const int t = threadIdx.x;
  const float ssq_w = ws[0];
  const float ssq_c = ws[1];
  float fs = sigmoidf(sqrtf(ssq_w));
  fs = fminf(fmaxf(fs, 0.0f), 1.0f);

  // forgetting gate: x = [0.01, 1e-4] -> Linear(2,8) -> ReLU -> Linear(8,1) -> sigmoid
  float acc = fgb2[0];
#pragma unroll
  for (int j = 0; j < 8; ++j) {
    float h = 0.01f * fgw1[2 * j] + 0.0001f * fgw1[2 * j + 1] + fgb1[j];
    acc += fmaxf(h, 0.0f) * fgw2[j];
  }
  const float gs = sigmoidf(acc);

  if (t < 16) {
    // hidden_bias[j] = fs*ur_w1[j,1] + gs*ur_w1[j,2] + ur_b1[j]
    ws[8 + t] = fs * urw1[3 * t + 1] + gs * urw1[3 * t + 2] + urb1[t];
  }
  if (t == 0) {
    ws[3] = fs;
    ws[4] = gs;
    out_scalars[0] = fs;             // final_strength
    out_scalars[1] = sqrtf(ssq_c);   // change_magnitude
    out_scalars[2] = gs;             // forgetting_strength
    out_scalars[4] = 0.5f;           // weight_stability
    out_scalars[5] = 0.0f;           // memory_strength
  }
}

// ---------------------------------------------------------------------------
// Stage 3: per-element consolidation MLP via V_WMMA_F32_16X16X4_F32.
// MM1: D[j,e] = a[j]*w[e] + hb[j]   (bias fused through the C operand)
// MM2: u[e]   = sum_j c[j]*relu(D)[j,e]  as 4 accumulating K=4 WMMAs whose
//      A-rows are identical -> u[e] readable from row 0 of the result.
// Layout shuffles between the C/D layout of MM1 and the B layout of MM2 are
// done with ds_swizzle half-wave swaps (SWAPX16).
// ---------------------------------------------------------------------------
#if __has_builtin(__builtin_amdgcn_wmma_f32_16x16x4_f32)
#define WC_HAVE_WMMA 1
#else
#define WC_HAVE_WMMA 0
#endif

__global__ void __launch_bounds__(256) wc_update(
    const float* __restrict__ cw, float* __restrict__ out,
    const float* __restrict__ urw1, const float* __restrict__ urw2,
    const float* __restrict__ urb2, float* ws, int n) {
  const int lane = (int)(threadIdx.x & 31);
  const int l16 = lane & 15;
  const bool hi = lane >= 16;
  const float b2 = urb2[0];

#if WC_HAVE_WMMA
  // A1: slopes a[j]=ur_w1[j,0] in K=0 column only.
  v2f A1;
  A1[0] = hi ? 0.0f : urw1[3 * l16 + 0];
  A1[1] = 0.0f;
  // C1[j,e] = hb[j]: VGPR r holds rows r (lanes 0-15) and r+8 (lanes 16-31).
  v8f C1;
#pragma unroll
  for (int r = 0; r < 8; ++r) C1[r] = ws[8 + r + (hi ? 8 : 0)];
  // A2 chunk t: every row = [c[4t], c[4t+1], c[4t+2], c[4t+3]] (A layout:
  // VGPR0 = K0 (lo lanes)/K2 (hi lanes); VGPR1 = K1/K3).
  v2f A2[4];
#pragma unroll
  for (int t = 0; t < 4; ++t) {
    A2[t][0] = urw2[4 * t + (hi ? 2 : 0)];
    A2[t][1] = urw2[4 * t + 1 + (hi ? 2 : 0)];
  }
  const v8f Z = {};
#else
  // Scalar fallback: preload slopes / bias / output weights into registers.
  float aj[16], hbj[16], cj[16];
#pragma unroll
  for (int j = 0; j < 16; ++j) {
    aj[j] = urw1[3 * j + 0];
    hbj[j] = ws[8 + j];
    cj[j] = urw2[j];
  }
#endif

  float sum_abs = 0.0f;
  const int stride = (int)(gridDim.x * blockDim.x);
  for (int idx = (int)(blockIdx.x * blockDim.x + threadIdx.x); idx < n;
       idx += stride) {
    const float w = cw[idx];
    float upre;

#if WC_HAVE_WMMA
    const float wsw = swz<16>(w);  // other half-wave's 16 elements
    float u[2];
#pragma unroll
    for (int tile = 0; tile < 2; ++tile) {
      // B1: element values in K=0 row.
      v2f B1;
      B1[0] = hi ? 0.0f : (tile ? wsw : w);
      B1[1] = 0.0f;
      v8f d = __builtin_amdgcn_wmma_f32_16x16x4_f32(
          false, A1, false, B1, (short)0, C1, false, false);
      v8f h;
#pragma unroll
      for (int r = 0; r < 8; ++r) h[r] = fmaxf(d[r], 0.0f);
      // Half-swapped copies: hs_r holds the opposite half's row data.
      float hs0 = swz<16>(h[0]), hs1 = swz<16>(h[1]);
      float hs2 = swz<16>(h[2]), hs3 = swz<16>(h[3]);
      float hs4 = swz<16>(h[4]), hs5 = swz<16>(h[5]);
      float hs6 = swz<16>(h[6]), hs7 = swz<16>(h[7]);

      v2f B2;
      v8f e;
      // chunk0: K=0..3 -> rows 0,1 (lo halves) / 2,3 (lo halves moved hi)
      B2[0] = hi ? hs2 : h[0];
      B2[1] = hi ? hs3 : h[1];
      e = __builtin_amdgcn_wmma_f32_16x16x4_f32(false, A2[0], false, B2,
                                                (short)0, Z, false, false);
      // chunk1: K=4..7 -> rows 4,5 / 6,7
      B2[0] = hi ? hs6 : h[4];
      B2[1] = hi ? hs7 : h[5];
      e = __builtin_amdgcn_wmma_f32_16x16x4_f32(false, A2[1], false, B2,
                                                (short)0, e, false, false);
      // chunk2: K=8..11 -> rows 8,9 (hi halves moved lo) / 10,11 (hi halves)
      B2[0] = hi ? h[2] : hs0;
      B2[1] = hi ? h[3] : hs1;
      e = __builtin_amdgcn_wmma_f32_16x16x4_f32(false, A2[2], false, B2,
                                                (short)0, e, false, false);
      // chunk3: K=12..15 -> rows 12,13 / 14,15
      B2[0] = hi ? h[6] : hs4;
      B2[1] = hi ? h[7] : hs5;
      e = __builtin_amdgcn_wmma_f32_16x16x4_f32(false, A2[3], false, B2,
                                                (short)0, e, false, false);
      // Row 0 of E = u[e] replicated in both half-waves.
      u[tile] = e[0];
    }
    upre = hi ? u[1] : u[0];
#else
    float acc = 0.0f;
#pragma unroll
    for (int j = 0; j < 16; ++j)
      acc += fmaxf(w * aj[j] + hbj[j], 0.0f) * cj[j];
    upre = acc;
#endif

    const float upd = tanhf(upre + b2) * CONSOLIDATION_RATE;
    const float nw = fminf(fmaxf(w + upd, -10.0f), 10.0f);
    __builtin_nontemporal_store(nw, out + idx);  // never re-read on chip
    sum_abs += fabsf(nw);
  }

  sum_abs = waveReduceSum(sum_abs);
  if (lane == 0) atomicAdd(&ws[2], sum_abs);
}

// ---------------------------------------------------------------------------
// Stage 4: consolidation quality from mean(|new_weights|).
// ---------------------------------------------------------------------------
__global__ void wc_finalize(const float* ws, float* out_scalars, float invn) {
  if (threadIdx.x == 0) {
    const float mean_abs = ws[2] * invn;
    const float dq =
        (mean_abs > 0.1f && mean_abs < 0.9f) ? 1.0f : mean_abs;
    out_scalars[3] = 0.5f * (0.5f + dq);  // (weight_stability + dq)/2
  }
}

extern "C" void kernel_launch(void* const* d_in, const int* in_sizes, int n_in,
                              void* d_out, int out_size, void* d_ws,
                              size_t ws_size, hipStream_t stream) {
  const float* cw = (const float*)d_in[0];
  const float* chg = (const float*)d_in[1];
  const float* urw1 = (const float*)d_in[2];
  const float* urb1 = (const float*)d_in[3];
  const float* urw2 = (const float*)d_in[4];
  const float* urb2 = (const float*)d_in[5];
  const float* fgw1 = (const float*)d_in[6];
  const float* fgb1 = (const float*)d_in[7];
  const float* fgw2 = (const float*)d_in[8];
  const float* fgb2 = (const float*)d_in[9];
  float* out = (float*)d_out;
  float* ws = (float*)d_ws;
  const int n = in_sizes[0];  // 4096*4096
  float* out_scalars = out + n;

  wc_init<<<1, 32, 0, stream>>>(ws);
  wc_reduce_sq<<<2048, 256, 0, stream>>>((const v4f*)cw, (const v4f*)chg, ws,
                                         n / 4);
  wc_scalars<<<1, 32, 0, stream>>>(urw1, urb1, fgw1, fgb1, fgw2, fgb2, ws,
                                   out_scalars);
  wc_update<<<4096, 256, 0, stream>>>(cw, out, urw1, urw2, urb2, ws, n);
  wc_finalize<<<1, 32, 0, stream>>>(ws, out_scalars, 1.0f / (float)n);
}